// StateNetwork_78572131713117
// MI455X (gfx1250) — compile-verified
//
#include <hip/hip_runtime.h>
#include <cstdint>

#define N_ENT   2000
#define GAT_EMB 64
#define HID     256
#define BATCH   8
#define ALPHA_S 0.2f
#define MASKV   9e-15f
#define XP      6016     // x-buffer row pitch (floats), 16B aligned
#define PPITCH  2048     // P' row pitch (floats)
#define KCH     12       // K chunks for FC (6000 = 12*500, 500 % 4 == 0)
#define KLEN    500

// ---- workspace layout (float offsets; total ~16.9 MB) ----
#define H4_OFF  0              // 2000 * float4  (h0,h1,h2,0)
#define S1_OFF  8000
#define S2_OFF  10000
#define M_OFF   12000
#define Q_OFF   14000
#define CS_OFF  16000          // [0..2]=colsum(h), [3]=max(s2)
#define XB_OFF  16064          // 16 rows * XP   (rows 8..15 zero)
#define PT_OFF  112320         // 12 * 2048 FC partials
#define PP_OFF  136896         // 2000 * PPITCH  P' matrix

typedef __attribute__((ext_vector_type(2))) float v2f;
typedef __attribute__((ext_vector_type(4))) float v4f;
typedef __attribute__((ext_vector_type(8))) float v8f;
typedef __attribute__((ext_vector_type(4))) int   v4i;

__device__ __forceinline__ float lrelu(float v) { return fmaxf(v, ALPHA_S * v); }

// ---- Stage A: h = emb@W, s1, s2, max(s2), colsum(h), M_i, Q_i (1 block) ----
__global__ void __launch_bounds__(256) prep_kernel(const float* __restrict__ emb,
                                                   const float* __restrict__ W,
                                                   const float* __restrict__ a,
                                                   float* __restrict__ ws) {
  __shared__ float  Wl[GAT_EMB * 3];
  __shared__ float  al[6];
  __shared__ float4 red[256];
  const int t = threadIdx.x;
  if (t < GAT_EMB * 3) Wl[t] = W[t];
  if (t < 6)           al[t] = a[t];
  __syncthreads();

  float4* h4 = (float4*)(ws + H4_OFF);
  float cs0 = 0.f, cs1 = 0.f, cs2 = 0.f, mx = -1e30f;
  for (int i = t; i < N_ENT; i += 256) {
    const float* er = emb + i * GAT_EMB;
    float h0 = 0.f, h1 = 0.f, h2 = 0.f;
#pragma unroll 8
    for (int k = 0; k < GAT_EMB; ++k) {
      float e = er[k];
      h0 = fmaf(e, Wl[3 * k + 0], h0);
      h1 = fmaf(e, Wl[3 * k + 1], h1);
      h2 = fmaf(e, Wl[3 * k + 2], h2);
    }
    float s1 = h0 * al[0] + h1 * al[1] + h2 * al[2];
    float s2 = h0 * al[3] + h1 * al[4] + h2 * al[5];
    h4[i] = make_float4(h0, h1, h2, 0.f);
    ws[S1_OFF + i] = s1;
    ws[S2_OFF + i] = s2;
    cs0 += h0; cs1 += h1; cs2 += h2; mx = fmaxf(mx, s2);
  }
  red[t] = make_float4(cs0, cs1, cs2, mx);
  __syncthreads();
  for (int off = 128; off > 0; off >>= 1) {
    if (t < off) {
      float4 o = red[t + off]; float4 m = red[t];
      m.x += o.x; m.y += o.y; m.z += o.z; m.w = fmaxf(m.w, o.w);
      red[t] = m;
    }
    __syncthreads();
  }
  if (t == 0) {
    ws[CS_OFF + 0] = red[0].x; ws[CS_OFF + 1] = red[0].y;
    ws[CS_OFF + 2] = red[0].z; ws[CS_OFF + 3] = red[0].w;
  }
  const float maxS2 = red[0].w;
  // M_i = upper bound on row max of masked logits (lrelu monotone) -> stable softmax shift
  for (int i = t; i < N_ENT; i += 256) {
    float M = fmaxf(lrelu(ws[S1_OFF + i] + maxS2), MASKV);
    ws[M_OFF + i] = M;
    ws[Q_OFF + i] = expf(MASKV - M);   // masked-entry softmax weight (pre-normalization)
  }
}

// ---- Stage B: P'[i,j] = exp(lrelu(s1_i + s2_j) - M_i) - Q_i  (16 MB, L2-resident) ----
__global__ void __launch_bounds__(256) ppbuild_kernel(float* __restrict__ ws) {
  const int i = blockIdx.y;
  const int j = blockIdx.x * 256 + threadIdx.x;
  if (j >= N_ENT) return;
  float v = ws[S1_OFF + i] + ws[S2_OFF + j];
  ws[PP_OFF + (size_t)i * PPITCH + j] = expf(lrelu(v) - ws[M_OFF + i]) - ws[Q_OFF + i];
}

__global__ void __launch_bounds__(256) zerox_kernel(float* __restrict__ ws) {
  int idx = blockIdx.x * 256 + threadIdx.x;
  if (idx < 16 * XP) ws[XB_OFF + idx] = 0.f;
}

// ---- Stage C: stream adj (128 MB, NT, b128/lane), fused mask+softmax+aggregate+ELU.
//      One wave per (b,i) row; h staged SoA in LDS (conflict-free b128 reads);
//      P' row hits L2 (16 MB resident, reused by all 8 batches). ----
__global__ void __launch_bounds__(256) attagg_kernel(const int* __restrict__ adj,
                                                     const float* __restrict__ wsc,
                                                     float* __restrict__ ws) {
  __shared__ __align__(16) float h0s[N_ENT];
  __shared__ __align__(16) float h1s[N_ENT];
  __shared__ __align__(16) float h2s[N_ENT];
  const float4* h4g = (const float4*)(wsc + H4_OFF);
  for (int k = threadIdx.x; k < N_ENT; k += 256) {
    float4 v = h4g[k];
    h0s[k] = v.x; h1s[k] = v.y; h2s[k] = v.z;
  }
  __syncthreads();

  const int wave = threadIdx.x >> 5, lane = threadIdx.x & 31;
  const int r = blockIdx.x * 8 + wave;          // flattened (b, i) in [0, 16000)
  const int i = r % N_ENT;
  const int b = r / N_ENT;

  const float Q     = wsc[Q_OFF + i];
  const v4i*  arow4 = (const v4i*)(adj + (size_t)r * N_ENT);           // 16B aligned
  const v4f*  prow4 = (const v4f*)(wsc + PP_OFF + (size_t)i * PPITCH); // 16B aligned

  float s = 0.f, f0 = 0.f, f1 = 0.f, f2 = 0.f;
  for (int g = lane; g < N_ENT / 4; g += 32) {   // 128 contiguous cols per wave-iter
    v4i mv = __builtin_nontemporal_load(arow4 + g);  // adj: streamed, no reuse
    v4f pv = prow4[g];
    v4f h0v = *(const v4f*)(h0s + 4 * g);
    v4f h1v = *(const v4f*)(h1s + 4 * g);
    v4f h2v = *(const v4f*)(h2s + 4 * g);
    float t0 = (mv.x > 0) ? pv.x : 0.f;
    float t1 = (mv.y > 0) ? pv.y : 0.f;
    float t2 = (mv.z > 0) ? pv.z : 0.f;
    float t3 = (mv.w > 0) ? pv.w : 0.f;
    s += (t0 + t1) + (t2 + t3);
    f0 = fmaf(t0, h0v.x, fmaf(t1, h0v.y, fmaf(t2, h0v.z, fmaf(t3, h0v.w, f0))));
    f1 = fmaf(t0, h1v.x, fmaf(t1, h1v.y, fmaf(t2, h1v.z, fmaf(t3, h1v.w, f1))));
    f2 = fmaf(t0, h2v.x, fmaf(t1, h2v.y, fmaf(t2, h2v.z, fmaf(t3, h2v.w, f2))));
  }
  for (int off = 16; off > 0; off >>= 1) {
    s  += __shfl_xor(s,  off, 32);
    f0 += __shfl_xor(f0, off, 32);
    f1 += __shfl_xor(f1, off, 32);
    f2 += __shfl_xor(f2, off, 32);
  }
  if (lane == 0) {
    float cs0 = wsc[CS_OFF], cs1 = wsc[CS_OFF + 1], cs2 = wsc[CS_OFF + 2];
    float denom = fmaf((float)N_ENT, Q, s);   // every column contributes >= Q
    float inv = 1.f / denom;
    float n0 = fmaf(Q, cs0, f0) * inv;
    float n1 = fmaf(Q, cs1, f1) * inv;
    float n2 = fmaf(Q, cs2, f2) * inv;
    float* xr = ws + XB_OFF + (size_t)b * XP + i * 3;
    xr[0] = n0 > 0.f ? n0 : expm1f(n0);       // ELU(alpha=1)
    xr[1] = n1 > 0.f ? n1 : expm1f(n1);
    xr[2] = n2 > 0.f ? n2 : expm1f(n2);
  }
}

// ---- Stage D: out = x @ fc1_w^T via V_WMMA_F32_16X16X4_F32.
//      M=16 (rows 8..15 zero), 16 N-tiles x 12 K-chunks, deterministic partials. ----
#if __has_builtin(__builtin_amdgcn_wmma_f32_16x16x4_f32)
__global__ void __launch_bounds__(32) fc_wmma_kernel(const float* __restrict__ w,
                                                     const float* __restrict__ wsc,
                                                     float* __restrict__ ws) {
  const int lane = threadIdx.x;
  const int nt = blockIdx.x;       // N tile (0..15)
  const int kc = blockIdx.y;       // K chunk (0..11)
  const int m = lane & 15, half = lane >> 4;

  // A (16x4 f32): lanes 0-15 hold K=c,c+1 ; lanes 16-31 hold K=c+2,c+3 (row m)
  const float* xrow = wsc + XB_OFF + (size_t)m * XP;
  // B (4x16 f32): lane's N column = m; same K split across half-waves
  const float* wrow = w + (size_t)(nt * 16 + m) * (N_ENT * 3);

  v8f acc = {};
  const int c0 = kc * KLEN;
  for (int c = c0; c < c0 + KLEN; c += 4) {
    int ca = c + 2 * half;
    v2f A = *(const v2f*)(xrow + ca);
    v2f B = *(const v2f*)(wrow + ca);
    acc = __builtin_amdgcn_wmma_f32_16x16x4_f32(false, A, false, B, (short)0, acc,
                                                false, false);
  }
  // D layout: VGPR r, lanes 0-15 -> (M=r, N=lane); only M=0..7 valid
  float* part = ws + PT_OFF + kc * (BATCH * HID) + nt * 16;
  if (lane < 16) {
#pragma unroll
    for (int rI = 0; rI < 8; ++rI) part[rI * HID + m] = acc[rI];
  }
}
#else
// Fallback (should not trigger on gfx1250): plain VALU FC partials
__global__ void __launch_bounds__(32) fc_wmma_kernel(const float* __restrict__ w,
                                                     const float* __restrict__ wsc,
                                                     float* __restrict__ ws) {
  const int lane = threadIdx.x;
  const int nt = blockIdx.x, kc = blockIdx.y;
  if (lane >= 16) return;
  const int n = nt * 16 + lane;
  const int c0 = kc * KLEN;
  for (int b = 0; b < 8; ++b) {
    float s = 0.f;
    for (int c = c0; c < c0 + KLEN; ++c)
      s = fmaf(wsc[XB_OFF + (size_t)b * XP + c], w[(size_t)n * (N_ENT * 3) + c], s);
    ws[PT_OFF + kc * 2048 + b * HID + n] = s;
  }
}
#endif

__global__ void __launch_bounds__(256) fc_reduce_kernel(const float* __restrict__ bias,
                                                        const float* __restrict__ wsc,
                                                        float* __restrict__ out) {
  int idx = blockIdx.x * 256 + threadIdx.x;
  if (idx >= BATCH * HID) return;
  float sum = bias[idx & (HID - 1)];
#pragma unroll
  for (int kc = 0; kc < KCH; ++kc) sum += wsc[PT_OFF + kc * 2048 + idx];
  out[idx] = sum;
}

extern "C" void kernel_launch(void* const* d_in, const int* in_sizes, int n_in,
                              void* d_out, int out_size, void* d_ws, size_t ws_size,
                              hipStream_t stream) {
  const float* emb  = (const float*)d_in[0];
  const float* W    = (const float*)d_in[1];
  const float* a    = (const float*)d_in[2];
  const float* fc1w = (const float*)d_in[3];
  const float* fc1b = (const float*)d_in[4];
  const int*   adj  = (const int*)d_in[5];
  float* ws  = (float*)d_ws;
  float* out = (float*)d_out;

  prep_kernel<<<1, 256, 0, stream>>>(emb, W, a, ws);
  zerox_kernel<<<(16 * XP + 255) / 256, 256, 0, stream>>>(ws);
  dim3 gb((N_ENT + 255) / 256, N_ENT);
  ppbuild_kernel<<<gb, 256, 0, stream>>>(ws);
  attagg_kernel<<<N_ENT, 256, 0, stream>>>(adj, ws, ws);   // 2000 blocks * 8 waves = 16000 rows
  dim3 gf(16, KCH);
  fc_wmma_kernel<<<gf, 32, 0, stream>>>(fc1w, ws, ws);
  fc_reduce_kernel<<<(BATCH * HID + 255) / 256, 256, 0, stream>>>(fc1b, ws, out);
}